// AccFlow_90168543412813
// MI455X (gfx1250) — compile-verified
//
#include <hip/hip_runtime.h>
#include <math.h>

// Vector types for WMMA operands (wave32):
// A (16x4 f32)  = 64 vals / 32 lanes = 2 VGPRs -> float2
// B (4x16 f32)  = 64 vals / 32 lanes = 2 VGPRs -> float2
// C/D (16x16 f32) = 256 vals / 32 lanes = 8 VGPRs -> float8
typedef __attribute__((ext_vector_type(2))) float v2f;
typedef __attribute__((ext_vector_type(8))) float v8f;

#define KNN_EPS 1e-8f
#define FLT_BIG 3.4e38f

// Branchless sorted insert into (t0<=t1<=t2) top-3 of smallest values.
// The three compare results are shared by value AND index selects, which
// keeps the backend on v_cndmask (no exec-mask if-conversion).
__device__ __forceinline__ void insert3(float c, int idx,
                                        float& t0, float& t1, float& t2,
                                        int& i0, int& i1, int& i2) {
    bool l2 = c < t2;
    bool l1 = c < t1;
    bool l0 = c < t0;
    float nt2 = l1 ? t1 : (l2 ? c : t2); int ni2 = l1 ? i1 : (l2 ? idx : i2);
    float nt1 = l0 ? t0 : (l1 ? c : t1); int ni1 = l0 ? i0 : (l1 ? idx : i1);
    float nt0 = l0 ? c  : t0;            int ni0 = l0 ? idx : i0;
    t0 = nt0; t1 = nt1; t2 = nt2;
    i0 = ni0; i1 = ni1; i2 = ni2;
}

// One block = 16 queries. 8 waves per block; each wave scans 1/8 of ref tiles.
// Per 16-ref tile: one V_WMMA_F32_16X16X4_F32 computes |r|^2 - 2*r.q for a
// 16(ref) x 16(query) tile. Lane L<16 holds query col L, refs m=0..7 in the
// 8 D-VGPRs; lane L>=16 holds the same query col, refs m=8..15.
__global__ __launch_bounds__(256) void knn_flow_wmma(
    const float* __restrict__ qp,   // [Nq,3]
    const float* __restrict__ rp,   // [Nr,3]
    const float* __restrict__ rf,   // [Nr,3]
    float* __restrict__ out,        // [Nq,3]
    int Nq, int Nr)
{
    __shared__ float s_d[8][16][3];
    __shared__ int   s_i[8][16][3];

    const int lane = threadIdx.x & 31;
    const int wave = threadIdx.x >> 5;
    const int hi   = lane >> 4;      // 0: refs m=0..7 / K=0,1 ; 1: refs m=8..15 / K=2,3
    const int l16  = lane & 15;

    const int qbase = blockIdx.x * 16;
    const int q     = qbase + l16;   // this lane's query column

    // Build B operand: column n = query; K0..K3 = (qx, qy, qz, 1)
    const float qx = qp[q * 3 + 0];
    const float qy = qp[q * 3 + 1];
    const float qz = qp[q * 3 + 2];
    v2f b;
    b.x = hi ? qz   : qx;   // VGPR0: K=0 (lo half) / K=2 (hi half)
    b.y = hi ? 1.0f : qy;   // VGPR1: K=1 (lo half) / K=3 (hi half)

    // Two independent top-3 accumulators (even / odd D registers) to halve
    // the serial insert dependency chain and feed the dual-issue VALU.
    float a0 = FLT_BIG, a1 = FLT_BIG, a2 = FLT_BIG;
    int   ai0 = 0, ai1 = 0, ai2 = 0;
    float b0 = FLT_BIG, b1 = FLT_BIG, b2 = FLT_BIG;
    int   bi0 = 0, bi1 = 0, bi2 = 0;

    const int ntiles = Nr >> 4;
    for (int t = wave; t < ntiles; t += 8) {
        // A operand: row m = ref; K0..K3 = (-2rx, -2ry, -2rz, |r|^2)
        const int r = t * 16 + l16;
        const float rx = rp[r * 3 + 0];
        const float ry = rp[r * 3 + 1];
        const float rz = rp[r * 3 + 2];
        const float r2 = rx * rx + ry * ry + rz * rz;
        v2f av;
        av.x = hi ? (-2.0f * rz) : (-2.0f * rx);  // VGPR0: K=0 / K=2
        av.y = hi ? r2           : (-2.0f * ry);  // VGPR1: K=1 / K=3

        v8f cz = {0.0f, 0.0f, 0.0f, 0.0f, 0.0f, 0.0f, 0.0f, 0.0f};
        // 8 args: (neg_a, A, neg_b, B, c_mod, C, reuse_a, reuse_b)
        v8f d = __builtin_amdgcn_wmma_f32_16x16x4_f32(
            false, av, false, b, (short)0, cz, false, false);

        // Wave-uniform skip: if no lane has any candidate below its current
        // 3rd-best, the 8 inserts are all no-ops -> branch over them.
        const float m = fminf(fminf(fminf(d[0], d[1]), fminf(d[2], d[3])),
                              fminf(fminf(d[4], d[5]), fminf(d[6], d[7])));
        if (__any(m < fmaxf(a2, b2))) {
            const int rb = t * 16 + (hi ? 8 : 0);
            insert3(d[0], rb + 0, a0, a1, a2, ai0, ai1, ai2);
            insert3(d[1], rb + 1, b0, b1, b2, bi0, bi1, bi2);
            insert3(d[2], rb + 2, a0, a1, a2, ai0, ai1, ai2);
            insert3(d[3], rb + 3, b0, b1, b2, bi0, bi1, bi2);
            insert3(d[4], rb + 4, a0, a1, a2, ai0, ai1, ai2);
            insert3(d[5], rb + 5, b0, b1, b2, bi0, bi1, bi2);
            insert3(d[6], rb + 6, a0, a1, a2, ai0, ai1, ai2);
            insert3(d[7], rb + 7, b0, b1, b2, bi0, bi1, bi2);
        }
    }

    // Merge odd accumulator into even.
    insert3(b0, bi0, a0, a1, a2, ai0, ai1, ai2);
    insert3(b1, bi1, a0, a1, a2, ai0, ai1, ai2);
    insert3(b2, bi2, a0, a1, a2, ai0, ai1, ai2);

    // Merge the two lane halves of each query column (lane L with lane L^16).
    {
        float u0 = __shfl_xor(a0, 16, 32);
        float u1 = __shfl_xor(a1, 16, 32);
        float u2 = __shfl_xor(a2, 16, 32);
        int   j0 = __shfl_xor(ai0, 16, 32);
        int   j1 = __shfl_xor(ai1, 16, 32);
        int   j2 = __shfl_xor(ai2, 16, 32);
        insert3(u0, j0, a0, a1, a2, ai0, ai1, ai2);
        insert3(u1, j1, a0, a1, a2, ai0, ai1, ai2);
        insert3(u2, j2, a0, a1, a2, ai0, ai1, ai2);
    }

    if (hi == 0) {
        s_d[wave][l16][0] = a0; s_d[wave][l16][1] = a1; s_d[wave][l16][2] = a2;
        s_i[wave][l16][0] = ai0; s_i[wave][l16][1] = ai1; s_i[wave][l16][2] = ai2;
    }
    __syncthreads();

    // Final cross-wave merge + output: one thread per query.
    if (threadIdx.x < 16) {
        const int n  = threadIdx.x;
        const int qq = qbase + n;

        float f0 = FLT_BIG, f1 = FLT_BIG, f2 = FLT_BIG;
        int   g0 = 0, g1 = 0, g2 = 0;
#pragma unroll
        for (int w = 0; w < 8; ++w) {
#pragma unroll
            for (int j = 0; j < 3; ++j) {
                insert3(s_d[w][n][j], s_i[w][n][j], f0, f1, f2, g0, g1, g2);
            }
        }

        const float cx = qp[qq * 3 + 0];
        const float cy = qp[qq * 3 + 1];
        const float cz = qp[qq * 3 + 2];
        const float q2 = cx * cx + cy * cy + cz * cz;

        const float d0 = sqrtf(fmaxf(f0 + q2, 0.0f));
        const float d1 = sqrtf(fmaxf(f1 + q2, 0.0f));
        const float d2 = sqrtf(fmaxf(f2 + q2, 0.0f));

        float w0 = 1.0f / (d0 + KNN_EPS);
        float w1 = 1.0f / (d1 + KNN_EPS);
        float w2 = 1.0f / (d2 + KNN_EPS);
        const float ws = w0 + w1 + w2;
        w0 /= ws; w1 /= ws; w2 /= ws;

#pragma unroll
        for (int c = 0; c < 3; ++c) {
            out[qq * 3 + c] = w0 * rf[g0 * 3 + c]
                            + w1 * rf[g1 * 3 + c]
                            + w2 * rf[g2 * 3 + c];
        }
    }
}

extern "C" void kernel_launch(void* const* d_in, const int* in_sizes, int n_in,
                              void* d_out, int out_size, void* d_ws, size_t ws_size,
                              hipStream_t stream) {
    (void)n_in; (void)d_ws; (void)ws_size; (void)out_size;
    const float* qp = (const float*)d_in[0];  // query_points [Nq,3]
    const float* rp = (const float*)d_in[1];  // ref_points   [Nr,3]
    const float* rf = (const float*)d_in[2];  // ref_flow     [Nr,3]
    float* out = (float*)d_out;               // [Nq,3]

    const int Nq = in_sizes[0] / 3;
    const int Nr = in_sizes[1] / 3;
    // k is fixed at 3 per the reference setup.

    const int grid = Nq / 16;  // one block per 16 queries
    knn_flow_wmma<<<grid, 256, 0, stream>>>(qp, rp, rf, out, Nq, Nr);
}